// MP_42494406427360
// MI455X (gfx1250) — compile-verified
//
#include <hip/hip_runtime.h>

// ---------------------------------------------------------------------------
// GNN message passing (fpa + bpa, K=3) for MI455X / gfx1250.
// - Pre-MLP hoisted to nodes: relu(FNN(y[src])) == relu(FNN(y))[src].
// - Each FNN (Linear-ReLU-Linear) is ONE fused kernel: layer-1 output lives in
//   LDS and feeds layer-2 directly (no HBM round-trip for the hidden layer).
// - The zero-sink + residual combine is fused into the update-FNN epilogue.
// - Matrix math: exact-fp32 V_WMMA_F32_16X16X4_F32 (matches reference numerics).
// ---------------------------------------------------------------------------

#define NN 10000
#define NE 320000
#define D  256
#define ROWS 32                 // rows per block (2 row-tiles of 16)
#define LDS_STRIDE 260          // 260 % 64 == 4 -> conflict-free b64 frag reads

typedef __attribute__((ext_vector_type(2))) float v2f;
typedef __attribute__((ext_vector_type(8))) float v8f;

__device__ __forceinline__ v8f wmma_f32(v2f a, v2f b, v8f c) {
    return __builtin_amdgcn_wmma_f32_16x16x4_f32(false, a, false, b,
                                                 (short)0, c, false, false);
}

// out = act(relu(A@W1+b1)@W2+b2), optionally fused with
// out[i, col_off+c] = (i==sink ? 0 : h[i,c]) + S[i,c]  (residual combine).
// Block: 256 threads = 8 waves; covers ROWS rows x 256 cols.
// Wave w owns cols [32w, 32w+32): 2 col-tiles x 2 row-tiles = 4 WMMA tiles.
__global__ __launch_bounds__(256) void fnn_kernel(
    const float* __restrict__ A,
    const float* __restrict__ W1, const float* __restrict__ bias1,
    const float* __restrict__ W2, const float* __restrict__ bias2,
    float* __restrict__ out,
    const float* __restrict__ S,     // nullptr -> no combine
    int relu_final, int sink, int out_stride, int col_off)
{
    __shared__ float lds[ROWS * LDS_STRIDE];   // A tile, then hidden tile
    const int t    = threadIdx.x;
    const int row0 = blockIdx.x * ROWS;

    // Coalesced cooperative load of the ROWSx256 A tile (rows clamped; the
    // clamp row NN-1 is owned by the clamping block itself -> in-place safe).
    #pragma unroll
    for (int idx = t; idx < ROWS * 256; idx += 256) {
        const int r = idx >> 8, c = idx & 255;
        int gr = row0 + r;
        if (gr >= NN) gr = NN - 1;
        lds[r * LDS_STRIDE + c] = A[(size_t)gr * D + c];
    }
    __syncthreads();

    const int wave = t >> 5;
    const int lane = t & 31;
    const int m    = lane & 15;   // M index (A frag) / N index (B, D frags)
    const int kh   = lane >> 4;   // k-half selector of the 16x4 f32 layout
    const int col0 = wave * 32;
    const int col1 = col0 + 16;

    // ---------------- layer 1: hidden = relu(A @ W1 + b1) ----------------
    v8f acc[2][2] = {};
    #pragma unroll 4
    for (int kk = 0; kk < 256; kk += 4) {
        const int krow = kk + 2 * kh;
        v2f a0, a1, bf0, bf1;
        a0.x = lds[m * LDS_STRIDE + krow];
        a0.y = lds[m * LDS_STRIDE + krow + 1];
        a1.x = lds[(16 + m) * LDS_STRIDE + krow];
        a1.y = lds[(16 + m) * LDS_STRIDE + krow + 1];
        bf0.x = W1[(size_t)krow * D + col0 + m];
        bf0.y = W1[(size_t)(krow + 1) * D + col0 + m];
        bf1.x = W1[(size_t)krow * D + col1 + m];
        bf1.y = W1[(size_t)(krow + 1) * D + col1 + m];
        acc[0][0] = wmma_f32(a0, bf0, acc[0][0]);
        acc[0][1] = wmma_f32(a0, bf1, acc[0][1]);
        acc[1][0] = wmma_f32(a1, bf0, acc[1][0]);
        acc[1][1] = wmma_f32(a1, bf1, acc[1][1]);
    }
    __syncthreads();   // all waves done reading the A tile

    // Write hidden tile (bias + ReLU) back into LDS.
    {
        const float bb0 = bias1[col0 + m];
        const float bb1 = bias1[col1 + m];
        #pragma unroll
        for (int rt = 0; rt < 2; ++rt) {
            #pragma unroll
            for (int v = 0; v < 8; ++v) {
                const int r = rt * 16 + v + 8 * kh;
                lds[r * LDS_STRIDE + col0 + m] = fmaxf(acc[rt][0][v] + bb0, 0.f);
                lds[r * LDS_STRIDE + col1 + m] = fmaxf(acc[rt][1][v] + bb1, 0.f);
            }
        }
    }
    __syncthreads();

    // ---------------- layer 2: out = hidden @ W2 + b2 ----------------
    v8f acc2[2][2] = {};
    #pragma unroll 4
    for (int kk = 0; kk < 256; kk += 4) {
        const int krow = kk + 2 * kh;
        v2f a0, a1, bf0, bf1;
        a0.x = lds[m * LDS_STRIDE + krow];
        a0.y = lds[m * LDS_STRIDE + krow + 1];
        a1.x = lds[(16 + m) * LDS_STRIDE + krow];
        a1.y = lds[(16 + m) * LDS_STRIDE + krow + 1];
        bf0.x = W2[(size_t)krow * D + col0 + m];
        bf0.y = W2[(size_t)(krow + 1) * D + col0 + m];
        bf1.x = W2[(size_t)krow * D + col1 + m];
        bf1.y = W2[(size_t)(krow + 1) * D + col1 + m];
        acc2[0][0] = wmma_f32(a0, bf0, acc2[0][0]);
        acc2[0][1] = wmma_f32(a0, bf1, acc2[0][1]);
        acc2[1][0] = wmma_f32(a1, bf0, acc2[1][0]);
        acc2[1][1] = wmma_f32(a1, bf1, acc2[1][1]);
    }

    // Epilogue: bias2 (+ReLU) (+zero-sink residual combine), guarded store.
    {
        const float bb0 = bias2[col0 + m];
        const float bb1 = bias2[col1 + m];
        #pragma unroll
        for (int rt = 0; rt < 2; ++rt) {
            #pragma unroll
            for (int v = 0; v < 8; ++v) {
                const int grow = row0 + rt * 16 + v + 8 * kh;
                if (grow < NN) {
                    float x0 = acc2[rt][0][v] + bb0;
                    float x1 = acc2[rt][1][v] + bb1;
                    if (relu_final) { x0 = fmaxf(x0, 0.f); x1 = fmaxf(x1, 0.f); }
                    if (S) {
                        if (grow == sink) { x0 = 0.f; x1 = 0.f; }
                        x0 += S[(size_t)grow * D + col0 + m];
                        x1 += S[(size_t)grow * D + col1 + m];
                    }
                    out[(size_t)grow * out_stride + col_off + col0 + m] = x0;
                    out[(size_t)grow * out_stride + col_off + col1 + m] = x1;
                }
            }
        }
    }
}

// Z[d[e],:] += P[s[e],:]  (segment-sum via global_atomic_add_f32).
// 4 edges per block, one thread per column -> coalesced gathers and atomics.
__global__ __launch_bounds__(256) void scatter_kernel(
    const float* __restrict__ P, const int* __restrict__ s_idx,
    const int* __restrict__ d_idx, float* __restrict__ Z)
{
    const int c  = threadIdx.x;
    const int e0 = blockIdx.x * 4;
    #pragma unroll
    for (int j = 0; j < 4; ++j) {
        const int e = e0 + j;
        const int s = s_idx[e];
        const int d = d_idx[e];
        atomicAdd(&Z[(size_t)d * D + c], P[(size_t)s * D + c]);
    }
}

extern "C" void kernel_launch(void* const* d_in, const int* in_sizes, int n_in,
                              void* d_out, int out_size, void* d_ws, size_t ws_size,
                              hipStream_t stream)
{
    (void)in_sizes; (void)n_in; (void)out_size; (void)ws_size;

    const float* x   = (const float*)d_in[0];
    const int*   src = (const int*)d_in[1];
    const int*   dst = (const int*)d_in[2];
    const float* ntW1 = (const float*)d_in[3];
    const float* ntb1 = (const float*)d_in[4];
    const float* ntW2 = (const float*)d_in[5];
    const float* ntb2 = (const float*)d_in[6];

    // Workspace: 3 buffers of [NN, 256] f32 (~31 MB total).
    const size_t buf = (size_t)NN * D;
    float* S = (float*)d_ws;      // self-transform (persists)
    float* Y = S + buf;           // current node state / pre-MLP output P
    float* Z = Y + buf;           // aggregated messages
    float* out = (float*)d_out;   // [NN, 512]

    const dim3 blk(256);
    const dim3 gFnn((NN + ROWS - 1) / ROWS);  // 313 blocks
    const dim3 gScat(NE / 4);                 // 80000 blocks

    // self_trans = relu(x@ntW1+b1)@ntW2+b2  (no outer relu, no combine)
    fnn_kernel<<<gFnn, blk, 0, stream>>>(x, ntW1, ntb1, ntW2, ntb2, S,
                                         nullptr, 0, -1, D, 0);

    struct Dir {
        const int *s, *d;
        int sink, col;
        const float *pW1, *pb1, *pW2, *pb2, *uW1, *ub1, *uW2, *ub2;
    };
    const Dir dirs[2] = {
        { src, dst, NN - 1, 0,
          (const float*)d_in[7],  (const float*)d_in[8],
          (const float*)d_in[9],  (const float*)d_in[10],
          (const float*)d_in[11], (const float*)d_in[12],
          (const float*)d_in[13], (const float*)d_in[14] },
        { dst, src, 0, D,
          (const float*)d_in[15], (const float*)d_in[16],
          (const float*)d_in[17], (const float*)d_in[18],
          (const float*)d_in[19], (const float*)d_in[20],
          (const float*)d_in[21], (const float*)d_in[22] },
    };

    for (int dd = 0; dd < 2; ++dd) {
        const Dir& g = dirs[dd];
        const float* y = S;  // first hop starts from the self-transform
        for (int k = 0; k < 3; ++k) {
            // P = relu(FNN_pre(y)) at nodes; Y->Y is in-place safe (each block
            // only writes rows it already consumed).
            fnn_kernel<<<gFnn, blk, 0, stream>>>(y, g.pW1, g.pb1, g.pW2, g.pb2,
                                                 Y, nullptr, 1, -1, D, 0);
            // Z = segment_sum(P[s], d)
            hipMemsetAsync(Z, 0, buf * sizeof(float), stream);
            scatter_kernel<<<gScat, blk, 0, stream>>>(Y, g.s, g.d, Z);
            // y' = relu(FNN_upd(Z)).at[sink].set(0) + S, fused epilogue.
            if (k < 2) {
                fnn_kernel<<<gFnn, blk, 0, stream>>>(Z, g.uW1, g.ub1, g.uW2, g.ub2,
                                                     Y, S, 1, g.sink, D, 0);
                y = Y;
            } else {
                fnn_kernel<<<gFnn, blk, 0, stream>>>(Z, g.uW1, g.ub1, g.uW2, g.ub2,
                                                     out, S, 1, g.sink, 2 * D, g.col);
            }
        }
    }
}